// Lorenz96_FS_7421703487574
// MI455X (gfx1250) — compile-verified
//
#include <hip/hip_runtime.h>
#include <cstdint>

// Lorenz96 two-scale RHS, f32, bandwidth-bound (~415 MB traffic -> ~18us floor @ 23.3 TB/s).
// Strategy: async-stage 8 contiguous rows (12672 B) per block into LDS via
// GLOBAL_LOAD_ASYNC_TO_LDS_B128 (CDNA5 async path, ASYNCcnt), compute cyclic
// stencils from LDS, write-once output with non-temporal stores.

#define XX   36
#define XY   10
#define ROW  396                    // XX*(XY+1)
#define RPB  8                      // rows per block (one per wave32)
#define BLK  256                    // 8 waves
#define CHUNKS ((ROW * RPB) / 4)    // 792 b128 chunks per block tile

__global__ __launch_bounds__(BLK)
void lorenz96_rhs_kernel(const float* __restrict__ u,
                         const float* __restrict__ param,
                         float* __restrict__ out,
                         int nrows) {
    __shared__ float sU[RPB * ROW]; // 12672 bytes

    const int tid = threadIdx.x;
    const long long rowBase = (long long)blockIdx.x * RPB;
    const float* gbase = u + rowBase * (long long)ROW;

    const bool full = (rowBase + RPB) <= (long long)nrows;   // block-uniform
    const int  nr   = full ? RPB : (int)((long long)nrows - rowBase);

    if (full) {
        // Flat async copy: 8 consecutive rows are contiguous in memory.
        for (int i = tid; i < CHUNKS; i += BLK) {
            unsigned           lds_off = (unsigned)(uintptr_t)(const void*)&sU[4 * i];
            unsigned long long gaddr   = (unsigned long long)(uintptr_t)(const void*)(gbase + 4 * i);
            asm volatile("global_load_async_to_lds_b128 %0, %1, off"
                         :: "v"(lds_off), "v"(gaddr)
                         : "memory");
        }
        asm volatile("s_wait_asynccnt 0" ::: "memory");
    } else {
        // Tail block (not hit for 131072 rows): plain guarded staging.
        for (int i = tid; i < nr * ROW; i += BLK) sU[i] = gbase[i];
    }
    __syncthreads();

    const float F = param[0];
    const float h = param[1];
    const float c = param[2];
    const float b = param[3];
    const float hc_m = h * c * (1.0f / (float)XY);  // h*c/10 (mean coefficient)
    const float h_m  = h * (1.0f / (float)XY);      // h/10

    const int wave = tid >> 5;   // row within block
    const int lane = tid & 31;

    if (wave < nr) {
        const float* row  = &sU[wave * ROW];
        float*       orow = out + (rowBase + wave) * (long long)ROW;

        for (int e = lane; e < ROW; e += 32) {
            float r;
            if (e < XX) {
                // dx[k] = (x[k+1]-x[k-2])*x[k-1] - x[k] + F - (h*c/10)*sum_j y[k,j]
                const int k   = e;
                const int kp1 = (k + 1 < XX) ? k + 1 : k + 1 - XX;
                const int km1 = (k >= 1) ? k - 1 : k - 1 + XX;
                const int km2 = (k >= 2) ? k - 2 : k - 2 + XX;
                const float* y = row + XX + k * XY;
                float s = 0.0f;
                #pragma unroll
                for (int j = 0; j < XY; ++j) s += y[j];
                r = (row[kp1] - row[km2]) * row[km1] - row[k] + F - hc_m * s;
            } else {
                // dy[k,j] = c*(-b*y[k,j+1]*(y[k,j+2]-y[k,j-1]) - y[k,j] + (h/10)*x[k])
                const int idx = e - XX;
                const int k   = idx / XY;
                const int j   = idx - k * XY;
                const int jp1 = (j + 1 < XY) ? j + 1 : j + 1 - XY;
                const int jp2 = (j + 2 < XY) ? j + 2 : j + 2 - XY;
                const int jm1 = (j >= 1) ? j - 1 : j - 1 + XY;
                const float* y = row + XX + k * XY;
                r = c * (-b * y[jp1] * (y[jp2] - y[jm1]) - y[j] + h_m * row[k]);
            }
            __builtin_nontemporal_store(r, &orow[e]);  // write-once stream, skip L2 retention
        }
    }
}

extern "C" void kernel_launch(void* const* d_in, const int* in_sizes, int n_in,
                              void* d_out, int out_size, void* d_ws, size_t ws_size,
                              hipStream_t stream) {
    // setup_inputs order: t (int scalar, unused), u (f32), param (4 x f32)
    const float* u     = (const float*)d_in[1];
    const float* param = (const float*)d_in[2];
    float*       out   = (float*)d_out;

    const int nrows   = in_sizes[1] / ROW;                 // 131072
    const int nblocks = (nrows + RPB - 1) / RPB;           // 16384

    lorenz96_rhs_kernel<<<dim3(nblocks), dim3(BLK), 0, stream>>>(u, param, out, nrows);
}